// BoundaryControlledMixer_11381663334581
// MI455X (gfx1250) — compile-verified
//
#include <hip/hip_runtime.h>
#include <math.h>

// ---------------- problem constants ----------------
constexpr int Bn = 2;
constexpr int Ln = 1024;
constexpr int DM = 768;
constexpr int NL = 4;
constexpr int En = 2 * DM;        // 1536
constexpr int Nn = 16;
constexpr int Kc = 4;             // conv width
constexpr int Rn = DM / 16;       // 48
constexpr int DBLW = Rn + 2 * Nn; // 80
constexpr int Mrows = Bn * Ln;    // 2048
constexpr int GPAD = 772;         // (DM+1) padded up to a multiple of 4

typedef float v2f __attribute__((ext_vector_type(2)));
typedef float v8f __attribute__((ext_vector_type(8)));

// ---------------- device helpers ----------------
__device__ __forceinline__ float sigm(float x) { return 1.0f / (1.0f + expf(-x)); }
__device__ __forceinline__ float silu_f(float x) { return x * sigm(x); }
__device__ __forceinline__ float softplus_f(float x) {
    return (x > 20.0f) ? x : log1pf(expf(x));
}

// =====================================================================
// Generic GEMM: out[m][n] = act( sum_k A[m][k] * W[n][k] + bias[n] )
// fp32 WMMA 16x16x4. Each wave owns a 16 x (16*NTW) strip: one A-fragment
// feeds NTW back-to-back WMMAs. The k-loop is software-pipelined (register
// double-buffering) so operand loads for step k+4 are in flight while the
// WMMAs for step k issue -> partial s_wait_loadcnt instead of full drains.
// REQUIREMENTS (enforced by callers): K % 4 == 0, K >= 8, lda/ldw even
// (=> all float2 operand loads are 8-byte aligned, no predication).
// act: 0=none 1=softplus 2=silu 3=sigmoid
// =====================================================================
template <int NTW>
__global__ void gemm_xwt_wmma(const float* __restrict__ A, int lda,
                              const float* __restrict__ W, int ldw,
                              float* __restrict__ C, int ldc,
                              int M, int K, int N,
                              const float* __restrict__ bias, int act) {
    const int wave = threadIdx.x >> 5;
    const int lane = threadIdx.x & 31;
    const int tile = blockIdx.x * (blockDim.x >> 5) + wave;
    const int ngroups = N / (16 * NTW);
    const int ntiles = (M >> 4) * ngroups;
    if (tile >= ntiles) return;   // wave-uniform exit: EXEC stays all-ones for WMMA

    const int mt = tile / ngroups;
    const int ng = tile - mt * ngroups;
    const int m0 = mt << 4;
    const int n0 = ng * (16 * NTW);
    const int half = lane >> 4;   // 0: K-pair {0,1}, 1: K-pair {2,3}
    const int lid  = lane & 15;

    const float* __restrict__ arow = A + (size_t)(m0 + lid) * lda + 2 * half;
    const float* __restrict__ wrow[NTW];
#pragma unroll
    for (int t = 0; t < NTW; ++t)
        wrow[t] = W + (size_t)(n0 + t * 16 + lid) * ldw + 2 * half;

    v8f acc[NTW];
#pragma unroll
    for (int t = 0; t < NTW; ++t) acc[t] = (v8f){};

    // prologue: fragments for k = 0
    v2f a_cur = *(const v2f*)(arow);
    v2f b_cur[NTW];
#pragma unroll
    for (int t = 0; t < NTW; ++t) b_cur[t] = *(const v2f*)(wrow[t]);

    // pipelined main loop: prefetch k+4 while multiplying k
    for (int k = 0; k + 4 < K; k += 4) {
        const v2f a_nxt = *(const v2f*)(arow + k + 4);
        v2f b_nxt[NTW];
#pragma unroll
        for (int t = 0; t < NTW; ++t) b_nxt[t] = *(const v2f*)(wrow[t] + k + 4);
#pragma unroll
        for (int t = 0; t < NTW; ++t) {
            acc[t] = __builtin_amdgcn_wmma_f32_16x16x4_f32(
                false, a_cur, false, b_cur[t], (short)0, acc[t], false, false);
        }
        a_cur = a_nxt;
#pragma unroll
        for (int t = 0; t < NTW; ++t) b_cur[t] = b_nxt[t];
    }
    // epilogue step: k = K-4
#pragma unroll
    for (int t = 0; t < NTW; ++t) {
        acc[t] = __builtin_amdgcn_wmma_f32_16x16x4_f32(
            false, a_cur, false, b_cur[t], (short)0, acc[t], false, false);
    }

#pragma unroll
    for (int t = 0; t < NTW; ++t) {
        const int nc = n0 + t * 16 + lid;
        const float bval = bias ? bias[nc] : 0.0f;
#pragma unroll
        for (int i = 0; i < 8; ++i) {
            const int row = m0 + i + 8 * half;  // VGPR i: lanes0-15 -> M=i, lanes16-31 -> M=i+8
            float v = acc[t][i] + bval;
            if (act == 1)      v = softplus_f(v);
            else if (act == 2) v = silu_f(v);
            else if (act == 3) v = sigm(v);
            C[(size_t)row * ldc + nc] = v;
        }
    }
}

// =====================================================================
// LayerNorm over last dim; one 256-thread block per row.
// =====================================================================
__global__ void layernorm_k(const float* __restrict__ in,
                            const float* __restrict__ w,
                            const float* __restrict__ b,
                            float* __restrict__ out, int dim) {
    const int row = blockIdx.x;
    const float* r = in + (size_t)row * dim;
    __shared__ float s1[256];
    __shared__ float s2[256];
    float sum = 0.0f, sq = 0.0f;
    for (int d = threadIdx.x; d < dim; d += blockDim.x) {
        const float v = r[d];
        sum += v; sq += v * v;
    }
    s1[threadIdx.x] = sum; s2[threadIdx.x] = sq;
    __syncthreads();
    for (int s = blockDim.x >> 1; s > 0; s >>= 1) {
        if (threadIdx.x < s) { s1[threadIdx.x] += s1[threadIdx.x + s]; s2[threadIdx.x] += s2[threadIdx.x + s]; }
        __syncthreads();
    }
    const float m   = s1[0] / (float)dim;
    const float var = s2[0] / (float)dim - m * m;
    const float inv = rsqrtf(var + 1e-5f);
    for (int d = threadIdx.x; d < dim; d += blockDim.x) {
        out[(size_t)row * dim + d] = (r[d] - m) * inv * w[d] + b[d];
    }
}

// =====================================================================
// Causal depthwise conv (K=4) + bias + SiLU.  Reads x-half of xz.
// =====================================================================
__global__ void conv_silu_k(const float* __restrict__ xz,
                            const float* __restrict__ cw,
                            const float* __restrict__ cb,
                            float* __restrict__ xc) {
    const int idx = blockIdx.x * blockDim.x + threadIdx.x;
    if (idx >= Bn * Ln * En) return;
    const int e = idx % En;
    const int l = (idx / En) % Ln;
    const int b = idx / (En * Ln);
    float acc = cb[e];
#pragma unroll
    for (int j = 0; j < Kc; ++j) {
        const int ll = l + j - (Kc - 1);
        if (ll >= 0)
            acc += cw[e * Kc + j] * xz[((size_t)(b * Ln + ll)) * (2 * En) + e];
    }
    xc[idx] = silu_f(acc);
}

// =====================================================================
// Selective-scan: one thread owns one (b,e) chain, N=16 state in regs.
// Fuses D-skip and z-gate: y = (scan + xt*D) * silu(z)
// =====================================================================
__global__ void scan_fused_k(const float* __restrict__ xc,
                             const float* __restrict__ dt,
                             const float* __restrict__ dbl,
                             const float* __restrict__ xz,
                             const float* __restrict__ A_log,
                             const float* __restrict__ ssm_D,
                             float* __restrict__ y) {
    const int tid = blockIdx.x * blockDim.x + threadIdx.x;
    if (tid >= Bn * En) return;
    const int b = tid / En;
    const int e = tid % En;

    float Ae[Nn], h[Nn];
#pragma unroll
    for (int n = 0; n < Nn; ++n) { Ae[n] = -expf(A_log[e * Nn + n]); h[n] = 0.0f; }
    const float Dskip = ssm_D[e];

    for (int l = 0; l < Ln; ++l) {
        const size_t row = (size_t)(b * Ln + l);
        const float xt  = xc[row * En + e];
        const float dtt = dt[row * En + e];
        const float* bl = dbl + row * DBLW;   // [48:64)=B [64:80)=C, wave-uniform (broadcast) loads
        const float dx  = dtt * xt;
        float ys = 0.0f;
#pragma unroll
        for (int n = 0; n < Nn; ++n) {
            const float dA = expf(dtt * Ae[n]);
            h[n] = dA * h[n] + dx * bl[Rn + n];
            ys  += h[n] * bl[Rn + Nn + n];
        }
        const float z = xz[row * (2 * En) + En + e];
        y[row * En + e] = (ys + xt * Dskip) * silu_f(z);
    }
}

// ---------------- small elementwise kernels ----------------
__global__ void copy_k(float* __restrict__ dst, const float* __restrict__ src, int n) {
    int i = blockIdx.x * blockDim.x + threadIdx.x;
    if (i < n) dst[i] = src[i];
}
__global__ void addinto_k(float* __restrict__ dst, const float* __restrict__ src, int n) {
    int i = blockIdx.x * blockDim.x + threadIdx.x;
    if (i < n) dst[i] += src[i];
}
// gin[m, 0:768] = x[m, :] ; gin[m, 768] = boundary_prob[m] ; gin[m, 769:772] = 0
__global__ void concat_pad_k(const float* __restrict__ x, const float* __restrict__ bp,
                             float* __restrict__ gin) {
    int i = blockIdx.x * blockDim.x + threadIdx.x;
    if (i >= Mrows * GPAD) return;
    const int m = i / GPAD;
    const int d = i - m * GPAD;
    float v = 0.0f;
    if (d < DM)        v = x[(size_t)m * DM + d];
    else if (d == DM)  v = bp[m];
    gin[i] = v;
}
// pad ctrl_w1 (768 x 769) -> (768 x 772), zeros in the tail columns
__global__ void padw1_k(const float* __restrict__ w, float* __restrict__ wp) {
    int i = blockIdx.x * blockDim.x + threadIdx.x;
    if (i >= DM * GPAD) return;
    const int n = i / GPAD;
    const int k = i - n * GPAD;
    wp[i] = (k < DM + 1) ? w[(size_t)n * (DM + 1) + k] : 0.0f;
}
// out = x + gate*(mixed - x)
__global__ void blend_k(const float* __restrict__ x, const float* __restrict__ gate,
                        const float* __restrict__ mixed, float* __restrict__ out, int n) {
    int i = blockIdx.x * blockDim.x + threadIdx.x;
    if (i < n) out[i] = x[i] + gate[i] * (mixed[i] - x[i]);
}

// =====================================================================
// Host-side orchestration
// =====================================================================
static inline int ceil_div(int a, int b) { return (a + b - 1) / b; }

extern "C" void kernel_launch(void* const* d_in, const int* in_sizes, int n_in,
                              void* d_out, int out_size, void* d_ws, size_t ws_size,
                              hipStream_t stream) {
    (void)in_sizes; (void)n_in; (void)out_size; (void)ws_size;
    const float* x          = (const float*)d_in[0];
    const float* bprob      = (const float*)d_in[1];
    const float* in_proj_w  = (const float*)d_in[2];
    const float* conv_w     = (const float*)d_in[3];
    const float* conv_b     = (const float*)d_in[4];
    const float* x_proj_w   = (const float*)d_in[5];
    const float* dt_proj_w  = (const float*)d_in[6];
    const float* dt_proj_b  = (const float*)d_in[7];
    const float* A_log      = (const float*)d_in[8];
    const float* ssm_D      = (const float*)d_in[9];
    const float* out_proj_w = (const float*)d_in[10];
    const float* ln_w       = (const float*)d_in[11];
    const float* ln_b       = (const float*)d_in[12];
    const float* normf_w    = (const float*)d_in[13];
    const float* normf_b    = (const float*)d_in[14];
    const float* ctrl_w1    = (const float*)d_in[15];
    const float* ctrl_b1    = (const float*)d_in[16];
    const float* ctrl_w2    = (const float*)d_in[17];
    const float* ctrl_b2    = (const float*)d_in[18];
    const float* out_ln_w   = (const float*)d_in[19];
    const float* out_ln_b   = (const float*)d_in[20];

    float* outp = (float*)d_out;                 // [0 : M*DM) = LN(out), [M*DM : 2*M*DM) = gate
    float* gate = outp + (size_t)Mrows * DM;

    // ---- workspace layout (floats) ----
    float* ws = (float*)d_ws;
    float* residual = ws;                                    // M*DM
    float* hs       = residual + (size_t)Mrows * DM;         // M*DM
    float* normed   = hs       + (size_t)Mrows * DM;         // M*DM
    float* xz       = normed   + (size_t)Mrows * DM;         // M*2E
    float* xc       = xz       + (size_t)Mrows * 2 * En;     // M*E
    float* dbl      = xc       + (size_t)Mrows * En;         // M*80
    float* dtb      = dbl      + (size_t)Mrows * DBLW;       // M*E
    float* ybuf     = dtb      + (size_t)Mrows * En;         // M*E
    // tail buffers alias the xz+xc region (layers done by then)
    float* gin      = xz;                                    // M*GPAD
    float* h1       = gin + (size_t)Mrows * GPAD;            // M*DM
    float* mixed    = h1  + (size_t)Mrows * DM;              // M*DM
    float* outb     = mixed + (size_t)Mrows * DM;            // M*DM
    float* w1pad    = outb  + (size_t)Mrows * DM;            // DM*GPAD

    const int EW = 256;
    auto launch_gemm = [&](const float* A, int lda, const float* W, int ldw,
                           float* C, int ldc, int M, int K, int N,
                           const float* bias, int act) {
        if (N % 64 == 0) {
            const int tiles = (M >> 4) * (N / 64);
            gemm_xwt_wmma<4><<<ceil_div(tiles, 8), 256, 0, stream>>>(
                A, lda, W, ldw, C, ldc, M, K, N, bias, act);
        } else if (N % 80 == 0) {
            const int tiles = (M >> 4) * (N / 80);
            gemm_xwt_wmma<5><<<ceil_div(tiles, 8), 256, 0, stream>>>(
                A, lda, W, ldw, C, ldc, M, K, N, bias, act);
        } else {
            const int tiles = (M >> 4) * (N >> 4);
            gemm_xwt_wmma<1><<<ceil_div(tiles, 8), 256, 0, stream>>>(
                A, lda, W, ldw, C, ldc, M, K, N, bias, act);
        }
    };

    const int nMD = Mrows * DM;

    // residual = x  (layer 0: residual is the input)
    copy_k<<<ceil_div(nMD, EW), EW, 0, stream>>>(residual, x, nMD);

    for (int i = 0; i < NL; ++i) {
        if (i > 0)  // residual = hs + residual
            addinto_k<<<ceil_div(nMD, EW), EW, 0, stream>>>(residual, hs, nMD);

        // normed = LN(residual, ln_w[i], ln_b[i])
        layernorm_k<<<Mrows, 256, 0, stream>>>(residual, ln_w + i * DM, ln_b + i * DM, normed, DM);

        // xz = normed @ in_proj_w[i]^T   (2048x768 x 768x3072)
        launch_gemm(normed, DM, in_proj_w + (size_t)i * 2 * En * DM, DM,
                    xz, 2 * En, Mrows, DM, 2 * En, nullptr, 0);

        // xc = silu(depthwise_causal_conv(xz[:, :E]) + conv_b)
        conv_silu_k<<<ceil_div(Bn * Ln * En, EW), EW, 0, stream>>>(
            xz, conv_w + (size_t)i * En * Kc, conv_b + (size_t)i * En, xc);

        // dbl = xc @ x_proj_w[i]^T       (2048x1536 x 1536x80)
        launch_gemm(xc, En, x_proj_w + (size_t)i * DBLW * En, En,
                    dbl, DBLW, Mrows, En, DBLW, nullptr, 0);

        // dt = softplus(dbl[:, :48] @ dt_proj_w[i]^T + dt_proj_b[i])  (2048x48 x 48x1536)
        launch_gemm(dbl, DBLW, dt_proj_w + (size_t)i * En * Rn, Rn,
                    dtb, En, Mrows, Rn, En, dt_proj_b + (size_t)i * En, /*softplus*/1);

        // selective scan + D-skip + z-gate  -> ybuf
        scan_fused_k<<<ceil_div(Bn * En, EW), EW, 0, stream>>>(
            xc, dtb, dbl, xz, A_log + (size_t)i * En * Nn, ssm_D + (size_t)i * En, ybuf);

        // hs = ybuf @ out_proj_w[i]^T    (2048x1536 x 1536x768)
        launch_gemm(ybuf, En, out_proj_w + (size_t)i * DM * En, En,
                    hs, DM, Mrows, En, DM, nullptr, 0);
    }

    // residual = hs + residual ; mixed = LN(residual, normf)
    addinto_k<<<ceil_div(nMD, EW), EW, 0, stream>>>(residual, hs, nMD);
    layernorm_k<<<Mrows, 256, 0, stream>>>(residual, normf_w, normf_b, mixed, DM);

    // controller: gin = [x, boundary_prob, 0-pad] (stride 772), w1 zero-padded to 768x772
    concat_pad_k<<<ceil_div(Mrows * GPAD, EW), EW, 0, stream>>>(x, bprob, gin);
    padw1_k<<<ceil_div(DM * GPAD, EW), EW, 0, stream>>>(ctrl_w1, w1pad);

    // h1 = silu(gin @ w1pad^T + ctrl_b1)      (K=772, guard-free)
    launch_gemm(gin, GPAD, w1pad, GPAD, h1, DM, Mrows, GPAD, DM, ctrl_b1, /*silu*/2);

    // gate = sigmoid(h1 @ ctrl_w2^T + ctrl_b2) -> directly into d_out second half
    launch_gemm(h1, DM, ctrl_w2, DM, gate, DM, Mrows, DM, DM, ctrl_b2, /*sigmoid*/3);

    // out = x + gate*(mixed - x) ; d_out[0:] = LN(out, out_ln)
    blend_k<<<ceil_div(nMD, EW), EW, 0, stream>>>(x, gate, mixed, outb, nMD);
    layernorm_k<<<Mrows, 256, 0, stream>>>(outb, out_ln_w, out_ln_b, outp, DM);
}